// Pointnet_FP_6227702580014
// MI455X (gfx1250) — compile-verified
//
#include <hip/hip_runtime.h>

typedef float v2f __attribute__((ext_vector_type(2)));
typedef float v8f __attribute__((ext_vector_type(8)));

#define B_    4
#define NT    8192
#define NS    2048
#define CS    256
#define CT    128
#define CIN   384          // Cs + Ct
#define NOUT  256
#define MTOT  (B_ * NT)    // 32768 rows

// ---------------------------------------------------------------------------
// Kernel 1: brute-force 3-NN with xyz_source staged in LDS (24 KB / 320 KB WGP)
// One block of 256 threads handles 256 target points of one batch.
// ---------------------------------------------------------------------------
__global__ __launch_bounds__(256) void three_nn_kernel(
    const float* __restrict__ xyz_t, const float* __restrict__ xyz_s,
    int* __restrict__ idx_out, float* __restrict__ w_out)
{
    __shared__ float sx[NS], sy[NS], sz[NS];
    const int blk  = blockIdx.x;       // 128 blocks total
    const int b    = blk >> 5;         // 32 blocks per batch
    const int tblk = blk & 31;

    const float* src = xyz_s + (size_t)b * NS * 3;
    for (int s = threadIdx.x; s < NS; s += 256) {
        sx[s] = src[s * 3 + 0];
        sy[s] = src[s * 3 + 1];
        sz[s] = src[s * 3 + 2];
    }
    __syncthreads();

    const size_t row = (size_t)b * NT + tblk * 256 + threadIdx.x;
    const float px = xyz_t[row * 3 + 0];
    const float py = xyz_t[row * 3 + 1];
    const float pz = xyz_t[row * 3 + 2];

    float d0 = 3.4e38f, d1 = 3.4e38f, d2 = 3.4e38f;
    int   i0 = 0, i1 = 0, i2 = 0;
    for (int s = 0; s < NS; ++s) {
        const float dx = px - sx[s];
        const float dy = py - sy[s];
        const float dz = pz - sz[s];
        const float d  = dx * dx + dy * dy + dz * dz;   // squared dist: order-preserving
        if (d < d2) {
            if (d < d1) {
                if (d < d0) { d2 = d1; i2 = i1; d1 = d0; i1 = i0; d0 = d; i0 = s; }
                else        { d2 = d1; i2 = i1; d1 = d;  i1 = s; }
            } else          { d2 = d;  i2 = s; }
        }
    }
    const float n0 = fmaxf(sqrtf(d0), 1e-10f);
    const float n1 = fmaxf(sqrtf(d1), 1e-10f);
    const float n2 = fmaxf(sqrtf(d2), 1e-10f);
    const float v0 = 1.0f / n0, v1 = 1.0f / n1, v2 = 1.0f / n2;
    const float inv_sum = 1.0f / (v0 + v1 + v2);

    idx_out[row * 3 + 0] = i0;
    idx_out[row * 3 + 1] = i1;
    idx_out[row * 3 + 2] = i2;
    w_out[row * 3 + 0] = v0 * inv_sum;
    w_out[row * 3 + 1] = v1 * inv_sum;
    w_out[row * 3 + 2] = v2 * inv_sum;
}

// ---------------------------------------------------------------------------
// Kernel 2: 3-point interpolate + concat -> xcat (MTOT x 384), coalesced in C
// One block (128 threads) per target row.
// ---------------------------------------------------------------------------
__global__ __launch_bounds__(128) void interp_concat_kernel(
    const float* __restrict__ feats_s, const float* __restrict__ feats_t,
    const int* __restrict__ idx, const float* __restrict__ wgt,
    float* __restrict__ xcat)
{
    const size_t row = blockIdx.x;
    const int b   = (int)(row >> 13);          // row / NT
    const int tid = threadIdx.x;

    const int   j0 = idx[row * 3 + 0];
    const int   j1 = idx[row * 3 + 1];
    const int   j2 = idx[row * 3 + 2];
    const float w0 = wgt[row * 3 + 0];
    const float w1 = wgt[row * 3 + 1];
    const float w2 = wgt[row * 3 + 2];

    const float* base = feats_s + (size_t)b * NS * CS;
    const float* r0 = base + (size_t)j0 * CS;
    const float* r1 = base + (size_t)j1 * CS;
    const float* r2 = base + (size_t)j2 * CS;

    // weights are normalized, so sum(w)=1; reference divides by (1 + EPS_WSUM)
    const float scale = 1.0f / (1.0f + 1e-6f);
    float* xr = xcat + row * (size_t)CIN;

    #pragma unroll
    for (int c = tid; c < CS; c += 128)
        xr[c] = (w0 * r0[c] + w1 * r1[c] + w2 * r2[c]) * scale;

    xr[CS + tid] = feats_t[row * (size_t)CT + tid];
}

// ---------------------------------------------------------------------------
// Kernel 3/4: fp32 GEMM + ReLU via V_WMMA_F32_16X16X4_F32.
// Block = 256 threads = 8 waves; each wave computes a 16x128 output tile
// (8 accumulators), reusing one A fragment across 8 WMMAs per K-step.
// Grid: (M/128, N/128).  N is fixed to 256.
// ---------------------------------------------------------------------------
__global__ __launch_bounds__(256) void gemm_relu_wmma(
    const float* __restrict__ A, const float* __restrict__ W,
    float* __restrict__ Out, int K)
{
    const int wave = threadIdx.x >> 5;
    const int lane = threadIdx.x & 31;
    const int half = lane >> 4;        // 0: lanes 0-15, 1: lanes 16-31
    const int l16  = lane & 15;
    const int mBase = blockIdx.x * 128 + wave * 16;
    const int nBase = blockIdx.y * 128;
    const int N = NOUT;

    v8f acc[8] = {};
    const float* arow = A + (size_t)(mBase + l16) * K;

    for (int k = 0; k < K; k += 4) {
        const int k0 = k + 2 * half;   // A lanes 0-15: K={k,k+1}; 16-31: K={k+2,k+3}
        const v2f a = *(const v2f*)(arow + k0);
        const float* wp = W + (size_t)k0 * N + nBase + l16;
        #pragma unroll
        for (int nt = 0; nt < 8; ++nt) {
            v2f bfrag;
            bfrag.x = wp[nt * 16];         // B row k0,   col nBase+nt*16+l16
            bfrag.y = wp[nt * 16 + N];     // B row k0+1, col ...
            acc[nt] = __builtin_amdgcn_wmma_f32_16x16x4_f32(
                /*neg_a=*/false, a, /*neg_b=*/false, bfrag,
                /*c_mod=*/(short)0, acc[nt],
                /*reuse_a=*/false, /*reuse_b=*/false);
        }
    }

    #pragma unroll
    for (int nt = 0; nt < 8; ++nt) {
        const int col = nBase + nt * 16 + l16;
        #pragma unroll
        for (int j = 0; j < 8; ++j) {
            const int r = mBase + j + 8 * half;   // C/D layout: VGPR j -> M=j (+8 hi-half)
            const float v = acc[nt][j];
            Out[(size_t)r * N + col] = v > 0.0f ? v : 0.0f;
        }
    }
}

// ---------------------------------------------------------------------------
extern "C" void kernel_launch(void* const* d_in, const int* in_sizes, int n_in,
                              void* d_out, int out_size, void* d_ws, size_t ws_size,
                              hipStream_t stream) {
    const float* xyz_target   = (const float*)d_in[0];
    const float* xyz_source   = (const float*)d_in[1];
    const float* feats_target = (const float*)d_in[2];
    const float* feats_source = (const float*)d_in[3];
    const float* w1           = (const float*)d_in[4];
    const float* w2           = (const float*)d_in[5];
    float* out = (float*)d_out;

    // workspace layout
    char* ws = (char*)d_ws;
    const size_t idx_bytes  = (size_t)MTOT * 3 * sizeof(int);    // 384 KB
    const size_t wgt_bytes  = (size_t)MTOT * 3 * sizeof(float);  // 384 KB
    const size_t xcat_bytes = (size_t)MTOT * CIN * sizeof(float);// 48 MB
    int*   idx  = (int*)ws;
    float* wgt  = (float*)(ws + idx_bytes);
    float* xcat = (float*)(ws + idx_bytes + wgt_bytes);
    float* hid  = (float*)(ws + idx_bytes + wgt_bytes + xcat_bytes);

    // 1) 3-NN (LDS-staged brute force)
    three_nn_kernel<<<dim3(B_ * (NT / 256)), dim3(256), 0, stream>>>(
        xyz_target, xyz_source, idx, wgt);

    // 2) interpolate + concat -> xcat (32768 x 384)
    interp_concat_kernel<<<dim3(MTOT), dim3(128), 0, stream>>>(
        feats_source, feats_target, idx, wgt, xcat);

    // 3) relu(xcat @ w1) -> hid (32768 x 256), K = 384
    gemm_relu_wmma<<<dim3(MTOT / 128, NOUT / 128), dim3(256), 0, stream>>>(
        xcat, w1, hid, CIN);

    // 4) relu(hid @ w2) -> out (32768 x 256), K = 256
    gemm_relu_wmma<<<dim3(MTOT / 128, NOUT / 128), dim3(256), 0, stream>>>(
        hid, w2, out, NOUT);
}